// FocusModules_90838558311175
// MI455X (gfx1250) — compile-verified
//
#include <hip/hip_runtime.h>

// ---------------------------------------------------------------------------
// FocusModules fused kernel for MI455X (gfx1250, wave32).
//
// Roofline: read x once (268 MB) + write compressed (268 MB) ~= 537 MB
// => ~23 us floor at 23.3 TB/s.  Compute (8192 windows x 32x32x256 Gram)
// ~= 4.3 GFLOP => fully memory-bound, so everything is fused into ONE pass:
// one wave per 32-token window
//   1. TDM tensor_load_to_lds DMAs rows [gbase-1 .. gbase+31] into LDS,
//      with hardware row padding (pad_interval=7/pad_amount=3 => 260-float
//      stride, 4-bank skew -> conflict-free WMMA fragment reads)
//   2. per-row reciprocal norms (eps-clamped like F.normalize)
//   3. stage-3 adjacent cosine mask
//   4. stage-1 Gram via V_WMMA_F32_16X16X4_F32 (exact f32 math so the
//      threshold decisions match the f32 reference)
//   5. row-mean -> mu/unbiased-sd -> keep mask (+argmin fallback)
//   6. non-temporal streaming store of compressed = x * mask3
// ---------------------------------------------------------------------------

typedef float        v2f  __attribute__((ext_vector_type(2)));
typedef float        v8f  __attribute__((ext_vector_type(8)));
typedef float        f4   __attribute__((ext_vector_type(4)));
typedef unsigned int u32x4 __attribute__((ext_vector_type(4)));
typedef int          i32x4 __attribute__((ext_vector_type(4)));
typedef int          i32x8 __attribute__((ext_vector_type(8)));

#define DIM    256
#define WIN    32
#define LDSTR  260   // padded LDS row stride (floats): 4-bank skew/row, 16B aligned

__global__ __launch_bounds__(32) void focus_fused(const float* __restrict__ x,
                                                  float* __restrict__ mask_out,
                                                  float* __restrict__ comp_out,
                                                  int N)
{
    __shared__ __align__(16) float xs[33 * LDSTR]; // row 0 = global row (gbase-1)
    __shared__ float rn[33];           // 1/max(||row||, 1e-12)
    __shared__ float ms[32];           // mean_sim per window row
    __shared__ float m3f[32];          // stage-3 keep mask per window row (0/1)

    const int tid   = threadIdx.x;     // 0..31, single wave32
    const int win   = blockIdx.x;
    const int gbase = win * WIN;

    // ---- Phase 1: fill LDS with rows [gbase-1 .. gbase+31] (row 0 = prev row)
    const int havePrev = (win > 0);
    const int nrows    = WIN + havePrev;
    const int row0     = 1 - havePrev;

#if __has_builtin(__builtin_amdgcn_tensor_load_to_lds)
    {
        // Tensor Data Mover: one DMA per wave moves the whole (nrows x 256) f32
        // tile into LDS and inserts 4 DWORDs of padding after every 256 DWORDs
        // (one row), producing the 260-float padded stride in hardware.
        const unsigned long long gaddr =
            (unsigned long long)(const void*)(x + ((long)gbase - havePrev) * DIM);
        const unsigned lds_addr =
            (unsigned)(unsigned long long)(const void*)&xs[row0 * LDSTR];

        u32x4 g0;
        g0.x = 0x1u;                                     // count=1, user mode
        g0.y = lds_addr;                                 // LDS byte address
        g0.z = (unsigned)(gaddr & 0xffffffffull);        // global_addr[31:0]
        g0.w = (unsigned)((gaddr >> 32) & 0x01ffffffull) // global_addr[56:32]
             | 0x80000000u;                              // type=2 ("image")

        i32x8 g1;
        g1[0] = (int)((2u << 16)       // data_size: 4-byte elements
                    | (1u << 20)       // pad_enable
                    | (7u << 22)       // pad_interval: 256 DWORDs (one row)
                    | (3u << 25));     // pad_amount: 4 DWORDs (16 B)
        g1[1] = (int)((DIM & 0xffffu) << 16);              // tensor_dim0[15:0]=256
        g1[2] = (int)(((unsigned)nrows & 0xffffu) << 16);  // dim0 hi=0 | tensor_dim1[15:0]
        g1[3] = (int)((unsigned)DIM << 16);                // dim1 hi=0 | tile_dim0=256
        g1[4] = (int)((unsigned)nrows & 0xffffu);          // tile_dim1=nrows, tile_dim2=0
        g1[5] = DIM;                                       // tensor_dim0_stride = 256
        g1[6] = 0;                                         // stride hi / dim1_stride lo
        g1[7] = 0;                                         // dim1_stride hi (unused, 2D)

        i32x4 z4 = (i32x4)0;
#if defined(__clang_major__) && (__clang_major__ >= 23)
        i32x8 z8 = (i32x8)0;
        __builtin_amdgcn_tensor_load_to_lds(g0, g1, z4, z4, z8, 0);
#else
        __builtin_amdgcn_tensor_load_to_lds(g0, g1, z4, z4, 0);
#endif
        __builtin_amdgcn_s_wait_tensorcnt((short)0);
    }
#else
    {
        const long gstart = (long)gbase - havePrev;
        const f4* src = (const f4*)(x + gstart * DIM);
        for (int i = tid; i < nrows * (DIM / 4); i += 32) {
            const int row = i >> 6;
            const int c4  = i & 63;
            *(f4*)&xs[(row0 + row) * LDSTR + (c4 << 2)] = src[i];
        }
    }
#endif
    if (!havePrev) {                     // synthesize zero "previous" row
        f4 z = (f4)0.0f;
        for (int i = tid; i < DIM / 4; i += 32)
            *(f4*)&xs[(i << 2)] = z;
    }
    __syncthreads();

    // ---- Phase 2: per-row reciprocal norms
    for (int r = tid; r < 33; r += 32) {
        const float* row = &xs[r * LDSTR];
        float s = 0.0f;
        #pragma unroll 8
        for (int c = 0; c < DIM; ++c) s += row[c] * row[c];
        rn[r] = 1.0f / fmaxf(sqrtf(s), 1e-12f);
    }
    __syncthreads();

    // ---- Phase 3: stage-3 adjacent cosine similarity -> keep mask
    {
        const float* ra = &xs[tid * LDSTR];         // global row gbase+tid-1
        const float* rb = &xs[(tid + 1) * LDSTR];   // global row gbase+tid
        float dot = 0.0f;
        #pragma unroll 8
        for (int c = 0; c < DIM; ++c) dot += ra[c] * rb[c];
        const float sim = dot * rn[tid] * rn[tid + 1];
        const bool keep3 = ((gbase + tid) == 0) ? true : (sim < 0.7f);
        m3f[tid] = keep3 ? 1.0f : 0.0f;
    }

    // ---- Phase 4: stage-1 Gram matrix G = Xn * Xn^T via f32 WMMA.
    // f32 16x16x4 fragment: lane L holds M(or N) = L&15, K = {0,1} (lanes 0-15)
    // / {2,3} (lanes 16-31) in (v.x, v.y).  A and B fragments over the same row
    // block are identical loads, so a0/a1 serve both roles.  Rows normalized on
    // the fly (xs stays raw so the final store writes exact x values).
    v8f c00 = (v8f)0.0f, c01 = (v8f)0.0f, c10 = (v8f)0.0f, c11 = (v8f)0.0f;
    {
        const int   mrow = tid & 15;
        const int   kof  = (tid >> 4) << 1;
        const float rnA  = rn[1 + mrow];             // window rows 0..15
        const float rnB  = rn[17 + mrow];            // window rows 16..31
        const float* pr0 = &xs[(1 + mrow) * LDSTR + kof];
        const float* pr1 = &xs[(17 + mrow) * LDSTR + kof];
        for (int k = 0; k < DIM; k += 4) {
            v2f a0, a1;
            a0.x = pr0[k] * rnA;  a0.y = pr0[k + 1] * rnA;
            a1.x = pr1[k] * rnB;  a1.y = pr1[k + 1] * rnB;
            c00 = __builtin_amdgcn_wmma_f32_16x16x4_f32(false, a0, false, a0, (short)0, c00, false, false);
            c01 = __builtin_amdgcn_wmma_f32_16x16x4_f32(false, a0, false, a1, (short)0, c01, false, false);
            c10 = __builtin_amdgcn_wmma_f32_16x16x4_f32(false, a1, false, a0, (short)0, c10, false, false);
            c11 = __builtin_amdgcn_wmma_f32_16x16x4_f32(false, a1, false, a1, (short)0, c11, false, false);
        }
    }

    // ---- Phase 5: row sums of G -> mean_sim -> mu / unbiased sd -> mask1.
    // D layout: VGPR r of lane L holds G[M][N], N = L&15, M = r (lanes 0-15) or
    // 8+r (lanes 16-31).  xor-shuffles 1,2,4,8 reduce within each 16-lane half.
    #pragma unroll
    for (int r = 0; r < 8; ++r) {
        float t = c00[r] + c01[r];                   // window rows 0..15
        t += __shfl_xor(t, 1); t += __shfl_xor(t, 2);
        t += __shfl_xor(t, 4); t += __shfl_xor(t, 8);
        float u = c10[r] + c11[r];                   // window rows 16..31
        u += __shfl_xor(u, 1); u += __shfl_xor(u, 2);
        u += __shfl_xor(u, 4); u += __shfl_xor(u, 8);
        if (tid == 0)  { ms[r]     = t * (1.0f / 32.0f); ms[16 + r] = u * (1.0f / 32.0f); }
        if (tid == 16) { ms[8 + r] = t * (1.0f / 32.0f); ms[24 + r] = u * (1.0f / 32.0f); }
    }
    __syncthreads();

    {
        const float v = ms[tid];
        float mu = v;
        mu += __shfl_xor(mu, 1); mu += __shfl_xor(mu, 2); mu += __shfl_xor(mu, 4);
        mu += __shfl_xor(mu, 8); mu += __shfl_xor(mu, 16);
        mu *= (1.0f / 32.0f);
        const float d = v - mu;
        float var = d * d;
        var += __shfl_xor(var, 1); var += __shfl_xor(var, 2); var += __shfl_xor(var, 4);
        var += __shfl_xor(var, 8); var += __shfl_xor(var, 16);
        var *= (1.0f / 31.0f);                        // torch .std() is unbiased
        const float sd = sqrtf(var);
        bool keep = !(v > mu + sd);

        unsigned long long bal = __ballot(keep);
        if ((bal & 0xffffffffull) == 0ull) {          // fallback: keep first argmin
            float mn = v;
            mn = fminf(mn, __shfl_xor(mn, 1));  mn = fminf(mn, __shfl_xor(mn, 2));
            mn = fminf(mn, __shfl_xor(mn, 4));  mn = fminf(mn, __shfl_xor(mn, 8));
            mn = fminf(mn, __shfl_xor(mn, 16));
            unsigned long long cand = __ballot(v == mn);
            const int first = __ffsll(cand) - 1;
            keep = (tid == first);
        }
        mask_out[gbase + tid] = keep ? 1.0f : 0.0f;
    }

    // ---- Phase 6: stream out compressed = x * mask3 (non-temporal: never re-read)
    {
        float* dst = comp_out + (long)gbase * DIM;
        for (int i = tid; i < WIN * (DIM / 4); i += 32) {
            const int row = i >> 6;
            const int c4  = i & 63;
            f4 v = *(const f4*)&xs[(row + 1) * LDSTR + (c4 << 2)];
            const float m = m3f[row];
            v *= m;
            __builtin_nontemporal_store(v, (f4*)(dst + (long)i * 4));
        }
    }
}

extern "C" void kernel_launch(void* const* d_in, const int* in_sizes, int n_in,
                              void* d_out, int out_size, void* d_ws, size_t ws_size,
                              hipStream_t stream) {
    const float* x = (const float*)d_in[0];
    const int N  = in_sizes[0] / DIM;      // 262144 tokens
    const int nw = N / WIN;                // 8192 windows
    float* mask_out = (float*)d_out;       // first N floats: mask1 (0/1)
    float* comp_out = (float*)d_out + N;   // then N*DIM floats: compressed
    hipLaunchKernelGGL(focus_fused, dim3(nw), dim3(32), 0, stream,
                       x, mask_out, comp_out, N);
}